// GridSample4D_35235911696910
// MI455X (gfx1250) — compile-verified
//
#include <hip/hip_runtime.h>

// GridSample4D on MI455X (gfx1250, wave32).
// inp [1,16,8,8,256,256] f32, grid [1,P,4] f32 -> out [1,16,P] f32.
//
// Gather-latency / L2-bound kernel:
//  - 8x 8-byte corner-pair gathers per channel per point (b64 path, align-4 ok on gfx1250)
//  - 16 effective weights (zeros-padding folded in) computed once per point
//  - global_prefetch_b8 overlaps channel-0 line fills with weight math
//  - NT hints on streaming grid loads + output stores keep the 192MB L2 for
//    the heavily reused input planes (~61 hits per 128B line per channel)

constexpr int C = 16, X = 8, Y = 8, U = 256, V = 256;
constexpr int PLANE = X * Y * U * V;           // 4,194,304 elements per channel
constexpr int PLANE_BYTES = PLANE * 4;         // 16 MB

// native clang vector (required by __builtin_nontemporal_load)
typedef float v4f __attribute__((ext_vector_type(4)));

// 4-byte-aligned float pair: (v, v+1) is only dword-aligned in general.
// gfx1250 executes b64 at 4B alignment (DWORD mode auto-aligns; UNALIGNED allows any).
struct __attribute__((packed, aligned(4))) f2 { float x, y; };

struct DimW { int i0c, i1c; float w0, w1; };

// align_corners=True unnormalize + floor/frac + zeros-padding weight masking
__device__ __forceinline__ DimW dim_prep(float g, int size) {
    float s = 0.5f * (float)(size - 1);
    float coord = fmaf(g, s, s);               // (g+1)*0.5*(size-1)
    float fl = floorf(coord);
    float f  = coord - fl;
    int i0 = (int)fl;
    int i1 = i0 + 1;
    DimW r;
    r.w0  = (i0 >= 0 && i0 < size) ? (1.0f - f) : 0.0f;
    r.w1  = (i1 >= 0 && i1 < size) ? f : 0.0f;
    r.i0c = min(max(i0, 0), size - 1);
    r.i1c = min(max(i1, 0), size - 1);
    return r;
}

// Innermost (contiguous) dim: pair loaded from vb = clamp(v0, 0, V-2);
// fold boundary selection into the two pair weights (av0 -> .x, av1 -> .y):
//   0 <= v0 <= V-2 : (1-f, f)       normal
//   v0 == -1       : (f, 0)         valid corner v=0 is element .x
//   v0 == V-1      : (0, 1-f)       valid corner v=V-1 is element .y
//   else           : (0, 0)         fully out of bounds
__device__ __forceinline__ void v_prep(float g, int size, int& vb, float& av0, float& av1) {
    float s = 0.5f * (float)(size - 1);
    float coord = fmaf(g, s, s);
    float fl = floorf(coord);
    float f  = coord - fl;
    int v0 = (int)fl;
    vb = min(max(v0, 0), size - 2);
    bool in_lo = (v0 >= 0) && (v0 <= size - 2);
    float w0 = 1.0f - f, w1 = f;
    av0 = in_lo ? w0 : ((v0 == -1)       ? w1 : 0.0f);
    av1 = in_lo ? w1 : ((v0 == size - 1) ? w0 : 0.0f);
}

__global__ __launch_bounds__(256)
void gs4d_kernel(const float* __restrict__ inp, const float* __restrict__ grid,
                 float* __restrict__ out, int P) {
    int p = blockIdx.x * blockDim.x + threadIdx.x;
    if (p >= P) return;

    // streaming, read-once: non-temporal 16B coalesced load
    v4f g = __builtin_nontemporal_load((const v4f*)(grid + 4 * (size_t)p));

    DimW dx = dim_prep(g.x, X);
    DimW dy = dim_prep(g.y, Y);
    DimW du = dim_prep(g.z, U);
    int vb; float av0, av1;
    v_prep(g.w, V, vb, av0, av1);

    // 8 pair byte-offsets within a channel plane, t = {x bit2, y bit1, u bit0}
    int rx0 = dx.i0c * (Y * U * V * 4), rx1 = dx.i1c * (Y * U * V * 4);
    int ry0 = dy.i0c * (U * V * 4),     ry1 = dy.i1c * (U * V * 4);
    int ru0 = du.i0c * (V * 4) + vb * 4, ru1 = du.i1c * (V * 4) + vb * 4;

    int off[8];
    off[0] = rx0 + ry0 + ru0;  off[1] = rx0 + ry0 + ru1;
    off[2] = rx0 + ry1 + ru0;  off[3] = rx0 + ry1 + ru1;
    off[4] = rx1 + ry0 + ru0;  off[5] = rx1 + ry0 + ru1;
    off[6] = rx1 + ry1 + ru0;  off[7] = rx1 + ry1 + ru1;

    const char* base = (const char*)inp;

    // gfx1250 global_prefetch_b8: start pulling channel-0 lines while the
    // weight products below are still in flight (lines are needed anyway).
#pragma unroll
    for (int t = 0; t < 8; ++t)
        __builtin_prefetch(base + off[t], 0, 3);

    // 16 effective pair weights, shared across all 16 channels
    float wxy00 = dx.w0 * dy.w0, wxy01 = dx.w0 * dy.w1;
    float wxy10 = dx.w1 * dy.w0, wxy11 = dx.w1 * dy.w1;
    float pw[8];
    pw[0] = wxy00 * du.w0;  pw[1] = wxy00 * du.w1;
    pw[2] = wxy01 * du.w0;  pw[3] = wxy01 * du.w1;
    pw[4] = wxy10 * du.w0;  pw[5] = wxy10 * du.w1;
    pw[6] = wxy11 * du.w0;  pw[7] = wxy11 * du.w1;
    float W0[8], W1[8];
#pragma unroll
    for (int t = 0; t < 8; ++t) { W0[t] = pw[t] * av0; W1[t] = pw[t] * av1; }

    // channel loop: 8 independent b64 gathers + 16 FMAs per channel;
    // unroll-by-4 keeps ~32 gathers outstanding per wave. Per-channel base is
    // uniform (SGPR bump), per-load address is SADDR + 32-bit VGPR offset.
#pragma unroll 4
    for (int c = 0; c < C; ++c) {
        const char* chan = base + (size_t)c * PLANE_BYTES;
        float acc = 0.0f;
#pragma unroll
        for (int t = 0; t < 8; ++t) {
            f2 pr = *(const f2*)(chan + off[t]);
            acc = fmaf(W0[t], pr.x, acc);
            acc = fmaf(W1[t], pr.y, acc);
        }
        // streaming, write-once: non-temporal coalesced store (keep L2 for input)
        __builtin_nontemporal_store(acc, out + (size_t)c * P + p);
    }
}

extern "C" void kernel_launch(void* const* d_in, const int* in_sizes, int n_in,
                              void* d_out, int out_size, void* d_ws, size_t ws_size,
                              hipStream_t stream) {
    const float* inp  = (const float*)d_in[0];
    const float* grid = (const float*)d_in[1];
    float* out = (float*)d_out;
    int P = in_sizes[1] / 4;            // grid is [1, P, 4]
    int threads = 256;                  // 8 wave32s per workgroup
    int blocks = (P + threads - 1) / threads;
    gs4d_kernel<<<blocks, threads, 0, stream>>>(inp, grid, out, P);
}